// DecoderRNN_14087492731348
// MI455X (gfx1250) — compile-verified
//
#include <hip/hip_runtime.h>
#include <hip/hip_bf16.h>

// Problem constants (match reference)
constexpr int Bk = 32, Pk = 196, Ek = 512, Hk = 512, Ak = 512, Vk = 32000, Tk = 50;

typedef __attribute__((ext_vector_type(2))) float v2f;
typedef __attribute__((ext_vector_type(8))) float v8f;

__device__ __forceinline__ v8f wmma4(v2f a, v2f b, v8f c) {
    // D = A(16x4 f32) * B(4x16 f32) + C(16x16 f32)
    return __builtin_amdgcn_wmma_f32_16x16x4_f32(false, a, false, b, (short)0, c,
                                                 false, false);
}

__device__ __forceinline__ float sigmoidf_(float x) {
    return 1.0f / (1.0f + __expf(-x));
}

// ---------------------------------------------------------------------------
// One 32(M) x 64(N) register tile worth of K-accumulation with compile-time
// strides: every load folds into base + 24-bit immediate offset, so the
// 6 loads per K=4 step can issue as one clause; 8 WMMAs per 6 loads.
// ---------------------------------------------------------------------------
template <int LDA, int LDW, int K>
__device__ __forceinline__ void wmma_tile_acc(
    const float* __restrict__ a0base,  // A row (m-subtile 0) + kk
    const float* __restrict__ w0base,  // W row (n-tile 0)    + kk
    v8f& c00, v8f& c10, v8f& c01, v8f& c11,
    v8f& c02, v8f& c12, v8f& c03, v8f& c13)
{
    #pragma unroll 4
    for (int k = 0; k < K; k += 4) {
        v2f av0 = *(const v2f*)(a0base + k);
        v2f av1 = *(const v2f*)(a0base + 16 * LDA + k);
        v2f bv0 = *(const v2f*)(w0base + k);
        v2f bv1 = *(const v2f*)(w0base + 16 * LDW + k);
        v2f bv2 = *(const v2f*)(w0base + 32 * LDW + k);
        v2f bv3 = *(const v2f*)(w0base + 48 * LDW + k);
        c00 = wmma4(av0, bv0, c00);
        c10 = wmma4(av1, bv0, c10);
        c01 = wmma4(av0, bv1, c01);
        c11 = wmma4(av1, bv1, c11);
        c02 = wmma4(av0, bv2, c02);
        c12 = wmma4(av1, bv2, c12);
        c03 = wmma4(av0, bv3, c03);
        c13 = wmma4(av1, bv3, c13);
    }
}

// ---------------------------------------------------------------------------
// Skinny GEMM: C[m][n] = sum_k A[m][k] * W[n][k] + bias[n]
// 4 waves / block; each wave computes a 32(M) x 64(N) register tile.
// Grids sized exactly -> EXEC all ones for every WMMA.
// ---------------------------------------------------------------------------
template <int LDA, int LDW, int K>
__global__ __launch_bounds__(128) void gemm_xt_wmma(
    const float* __restrict__ A,
    const float* __restrict__ W,
    const float* __restrict__ bias,
    float* __restrict__ C, int ldc)
{
    const int lane = threadIdx.x & 31;
    const int nt   = (blockIdx.x * 4 + (threadIdx.x >> 5)) * 4;  // four N tiles
    const int mt   = blockIdx.y;
    const int r    = lane & 15;
    const int kk   = (lane >> 4) << 1;   // 0 or 2

    const float* a0 = A + (size_t)(mt * 32 + r) * LDA + kk;
    const float* w0 = W + (size_t)(nt * 16 + r) * LDW + kk;

    v8f c00 = {}; v8f c10 = {};
    v8f c01 = {}; v8f c11 = {};
    v8f c02 = {}; v8f c12 = {};
    v8f c03 = {}; v8f c13 = {};
    wmma_tile_acc<LDA, LDW, K>(a0, w0, c00, c10, c01, c11, c02, c12, c03, c13);

    const int mb = mt * 32 + ((lane >> 4) << 3);
    #pragma unroll
    for (int j = 0; j < 4; ++j) {
        const int   n  = nt * 16 + j * 16 + r;
        const float bb = bias ? bias[n] : 0.0f;
        const v8f&  cA = (j == 0) ? c00 : (j == 1) ? c01 : (j == 2) ? c02 : c03;
        const v8f&  cB = (j == 0) ? c10 : (j == 1) ? c11 : (j == 2) ? c12 : c13;
        #pragma unroll
        for (int i = 0; i < 8; ++i) {
            C[(size_t)(mb + i) * ldc + n]      = cA[i] + bb;
            C[(size_t)(mb + i + 16) * ldc + n] = cB[i] + bb;
        }
    }
}

// ---------------------------------------------------------------------------
// Fused gate GEMM: gates[b][n] = x_t.Wih[:, :E] + awe.Wih[:, E:] + h.Whh + b
// M = 32 fixed, N = 4H = 2048, three constant-stride K=512 segments.
// ---------------------------------------------------------------------------
__global__ __launch_bounds__(128) void gates_wmma(
    const float* __restrict__ x0,   // embs + t*E, row stride Tk*Ek
    const float* __restrict__ awe,  // [B, E]
    const float* __restrict__ h,    // [B, H]
    const float* __restrict__ W_ih, // [4H, E+H]
    const float* __restrict__ W_hh, // [4H, H]
    const float* __restrict__ b_ih,
    const float* __restrict__ b_hh,
    float* __restrict__ gates)      // [B, 4H]
{
    const int lane = threadIdx.x & 31;
    const int nt   = (blockIdx.x * 4 + (threadIdx.x >> 5)) * 4;
    const int r    = lane & 15;
    const int kk   = (lane >> 4) << 1;

    v8f c00 = {}; v8f c10 = {};
    v8f c01 = {}; v8f c11 = {};
    v8f c02 = {}; v8f c12 = {};
    v8f c03 = {}; v8f c13 = {};

    // segment 1: x_t * W_ih[:, 0:E]
    wmma_tile_acc<Tk * Ek, Ek + Hk, Ek>(
        x0 + (size_t)r * (Tk * Ek) + kk,
        W_ih + (size_t)(nt * 16 + r) * (Ek + Hk) + kk,
        c00, c10, c01, c11, c02, c12, c03, c13);
    // segment 2: awe * W_ih[:, E:E+H]
    wmma_tile_acc<Ek, Ek + Hk, Hk>(
        awe + (size_t)r * Ek + kk,
        W_ih + (size_t)(nt * 16 + r) * (Ek + Hk) + Ek + kk,
        c00, c10, c01, c11, c02, c12, c03, c13);
    // segment 3: h * W_hh
    wmma_tile_acc<Hk, Hk, Hk>(
        h + (size_t)r * Hk + kk,
        W_hh + (size_t)(nt * 16 + r) * Hk + kk,
        c00, c10, c01, c11, c02, c12, c03, c13);

    const int mb = (lane >> 4) << 3;
    #pragma unroll
    for (int j = 0; j < 4; ++j) {
        const int   n  = nt * 16 + j * 16 + r;
        const float bb = b_ih[n] + b_hh[n];
        const v8f&  cA = (j == 0) ? c00 : (j == 1) ? c01 : (j == 2) ? c02 : c03;
        const v8f&  cB = (j == 0) ? c10 : (j == 1) ? c11 : (j == 2) ? c12 : c13;
        #pragma unroll
        for (int i = 0; i < 8; ++i) {
            gates[(size_t)(mb + i) * (4 * Hk) + n]      = cA[i] + bb;
            gates[(size_t)(mb + i + 16) * (4 * Hk) + n] = cB[i] + bb;
        }
    }
}

// ---------------------------------------------------------------------------
// Embedding gather: embs[b][t][e] = emb[captions[b][t]][e]
// ---------------------------------------------------------------------------
__global__ __launch_bounds__(256) void embed_gather(
    const float* __restrict__ emb, const int* __restrict__ caps,
    float* __restrict__ embs, int n)
{
    int i = blockIdx.x * 256 + threadIdx.x;
    if (i < n) {
        int e  = i & (Ek - 1);
        int bt = i >> 9;                 // / Ek
        embs[i] = emb[(size_t)caps[bt] * Ek + e];
    }
}

// ---------------------------------------------------------------------------
// scores[b][p] = sum_a relu(enc_att[b][p][a] + dec_att[b][a]) * w_full[a] + b_full
// One wave32 per (b,p) dot product, lane-shuffle reduction.
// ---------------------------------------------------------------------------
__global__ __launch_bounds__(256) void scores_kernel(
    const float* __restrict__ enc_att, const float* __restrict__ dec_att,
    const float* __restrict__ w_full, const float* __restrict__ b_full,
    float* __restrict__ scores)
{
    int wid  = (blockIdx.x * 256 + threadIdx.x) >> 5;
    int lane = threadIdx.x & 31;
    if (wid >= Bk * Pk) return;
    int b = wid / Pk;
    const float* ea = enc_att + (size_t)wid * Ak;
    const float* da = dec_att + (size_t)b * Ak;
    float acc = 0.0f;
    for (int a = lane; a < Ak; a += 32) {
        float v = ea[a] + da[a];
        v = v > 0.0f ? v : 0.0f;
        acc += v * w_full[a];
    }
    #pragma unroll
    for (int off = 16; off > 0; off >>= 1) acc += __shfl_xor(acc, off, 32);
    if (lane == 0) scores[wid] = acc + b_full[0];
}

// ---------------------------------------------------------------------------
// Per-batch-row softmax over P, then awe[b][e] = sum_p alpha[p] * features[b][p][e]
// ---------------------------------------------------------------------------
__global__ __launch_bounds__(256) void softmax_awe_kernel(
    const float* __restrict__ scores, const float* __restrict__ features,
    float* __restrict__ awe)
{
    __shared__ float sh[256];
    __shared__ float alpha[Pk];
    const int b = blockIdx.x, tid = threadIdx.x;

    float v = (tid < Pk) ? scores[b * Pk + tid] : -3.0e38f;
    sh[tid] = v;
    __syncthreads();
    for (int s = 128; s > 0; s >>= 1) {
        if (tid < s) sh[tid] = fmaxf(sh[tid], sh[tid + s]);
        __syncthreads();
    }
    const float mx = sh[0];
    __syncthreads();
    float e = (tid < Pk) ? __expf(v - mx) : 0.0f;
    sh[tid] = e;
    __syncthreads();
    for (int s = 128; s > 0; s >>= 1) {
        if (tid < s) sh[tid] += sh[tid + s];
        __syncthreads();
    }
    const float inv = 1.0f / sh[0];
    if (tid < Pk) alpha[tid] = e * inv;
    __syncthreads();

    const float* fb = features + (size_t)b * Pk * Ek;
    for (int ec = tid; ec < Ek; ec += 256) {
        float acc = 0.0f;
        for (int p = 0; p < Pk; ++p) acc += alpha[p] * fb[(size_t)p * Ek + ec];
        awe[(size_t)b * Ek + ec] = acc;
    }
}

// ---------------------------------------------------------------------------
// LSTM pointwise (torch gate order i,f,g,o); h,c updated in place.
// ---------------------------------------------------------------------------
__global__ __launch_bounds__(256) void lstm_kernel(
    const float* __restrict__ gates, float* __restrict__ h, float* __restrict__ c)
{
    int i = blockIdx.x * 256 + threadIdx.x;     // over B*H
    if (i >= Bk * Hk) return;
    int b = i / Hk, n = i - b * Hk;
    const float* g = gates + (size_t)b * 4 * Hk;
    float ig = sigmoidf_(g[n]);
    float fg = sigmoidf_(g[Hk + n]);
    float gg = tanhf(g[2 * Hk + n]);
    float og = sigmoidf_(g[3 * Hk + n]);
    float cn = fg * c[i] + ig * gg;
    c[i] = cn;
    h[i] = og * tanhf(cn);
}

__global__ __launch_bounds__(256) void zero2_kernel(float* a, float* b, int n)
{
    int i = blockIdx.x * 256 + threadIdx.x;
    if (i < n) { a[i] = 0.0f; b[i] = 0.0f; }
}

// ---------------------------------------------------------------------------
extern "C" void kernel_launch(void* const* d_in, const int* in_sizes, int n_in,
                              void* d_out, int out_size, void* d_ws, size_t ws_size,
                              hipStream_t stream)
{
    const float* features = (const float*)d_in[0];
    const int*   captions = (const int*)  d_in[1];
    const float* emb      = (const float*)d_in[2];
    const float* W_ih     = (const float*)d_in[3];
    const float* b_ih     = (const float*)d_in[4];
    const float* W_hh     = (const float*)d_in[5];
    const float* b_hh     = (const float*)d_in[6];
    const float* W_enc    = (const float*)d_in[7];
    const float* b_enc    = (const float*)d_in[8];
    const float* W_dec    = (const float*)d_in[9];
    const float* b_dec    = (const float*)d_in[10];
    const float* W_full   = (const float*)d_in[11];
    const float* b_full   = (const float*)d_in[12];
    const float* W_out    = (const float*)d_in[13];
    const float* b_out    = (const float*)d_in[14];
    float* out = (float*)d_out;

    float* ws      = (float*)d_ws;
    float* embs    = ws; ws += (size_t)Bk * Tk * Ek;   // 819200
    float* enc_att = ws; ws += (size_t)Bk * Pk * Ak;   // 3211264
    float* dec_att = ws; ws += (size_t)Bk * Ak;
    float* scores  = ws; ws += (size_t)Bk * Pk;        // 6272 (even)
    float* awe     = ws; ws += (size_t)Bk * Ek;
    float* gates   = ws; ws += (size_t)Bk * 4 * Hk;
    float* hbuf    = ws; ws += (size_t)Bk * Hk;
    float* cbuf    = ws; ws += (size_t)Bk * Hk;

    // h0 = c0 = 0 (ws is poisoned; must init every call)
    zero2_kernel<<<(Bk * Hk + 255) / 256, 256, 0, stream>>>(hbuf, cbuf, Bk * Hk);

    // Embedding lookup for all timesteps
    embed_gather<<<(Bk * Tk * Ek + 255) / 256, 256, 0, stream>>>(
        emb, captions, embs, Bk * Tk * Ek);

    // enc_att[b,p,:] = features[b,p,:] @ W_enc^T + b_enc   (M=6272, N=512, K=512)
    gemm_xt_wmma<Ek, Ek, Ek><<<dim3(Ak / 64 / 4, (Bk * Pk) / 32), 128, 0, stream>>>(
        features, W_enc, b_enc, enc_att, Ak);

    for (int t = 0; t < Tk; ++t) {
        // dec_att = h @ W_dec^T + b_dec   (M=32, N=512, K=512)
        gemm_xt_wmma<Hk, Hk, Hk><<<dim3(Ak / 64 / 4, 1), 128, 0, stream>>>(
            hbuf, W_dec, b_dec, dec_att, Ak);

        // scores = relu(enc_att + dec_att) . W_full + b_full
        scores_kernel<<<(Bk * Pk * 32) / 256, 256, 0, stream>>>(
            enc_att, dec_att, W_full, b_full, scores);

        // alpha = softmax(scores); awe = alpha . features
        softmax_awe_kernel<<<Bk, 256, 0, stream>>>(scores, features, awe);

        // gates = [x_t, awe] @ W_ih^T + h @ W_hh^T + b  (M=32, N=2048)
        gates_wmma<<<dim3((4 * Hk) / 64 / 4, 1), 128, 0, stream>>>(
            embs + (size_t)t * Ek, awe, hbuf, W_ih, W_hh, b_ih, b_hh, gates);

        // LSTM pointwise, in-place h/c update
        lstm_kernel<<<(Bk * Hk + 255) / 256, 256, 0, stream>>>(gates, hbuf, cbuf);

        // out[:, t, :] = h @ W_out^T + b_out   (M=32, N=32000, K=512)
        gemm_xt_wmma<Hk, Hk, Hk><<<dim3(Vk / 64 / 4, 1), 128, 0, stream>>>(
            hbuf, W_out, b_out, out + (size_t)t * Vk, Tk * Vk);
    }
}